// GammatoneFilterBankAnalysis_77635828842574
// MI455X (gfx1250) — compile-verified
//
#include <hip/hip_runtime.h>

// ---------------------------------------------------------------------------
// Gammatone filterbank (order-4 complex IIR) on gfx1250.
//
//   y[t] = K*x[t] - sum_{p=1..4} a_p * y[t-p]        (complex, per channel)
//
// Block-of-16 formulation (exact):
//   Y(16t x 16col) = HKr*Xe + HKi*Xo + Fr*S + Fi*S~  via v_wmma_f32_16x16x4_f32
// Columns = 8 batches x {re, im}.  HK = gain * lower-tri Toeplitz of impulse
// response h; F = homogeneous state response; S~ = col-pair swap+negate
// (complex multiply trick).  Time parallelized with a 3-phase chunked scan.
// x staging is double-buffered through LDS so global-load latency overlaps
// the WMMA chain; output stores are non-temporal (no reuse, 307 MB stream).
// ---------------------------------------------------------------------------

typedef __attribute__((ext_vector_type(2))) float v2f;
typedef __attribute__((ext_vector_type(8))) float v8f;

struct cpx { float r, i; };
__device__ __forceinline__ cpx cmul(cpx a, cpx b){ return {a.r*b.r - a.i*b.i, a.r*b.i + a.i*b.r}; }
__device__ __forceinline__ cpx cadd(cpx a, cpx b){ return {a.r+b.r, a.i+b.i}; }
__device__ __forceinline__ cpx csub(cpx a, cpx b){ return {a.r-b.r, a.i-b.i}; }

#define WMMA_F32_16x16x4(A,B,C) \
  __builtin_amdgcn_wmma_f32_16x16x4_f32(false,(A),false,(B),(short)0,(C),false,false)

// One wave (32 lanes) per (channel k, chunk c).  STORE=false: state-only pass
// writing chunk-final states p_c.  STORE=true: full pass with stitched
// incoming state, writing outputs.
template <bool STORE>
__global__ __launch_bounds__(32) void gt_chunk(
    const float* __restrict__ x, const float* __restrict__ acoef,
    const float* __restrict__ sin_ws, float* __restrict__ p_ws,
    float* __restrict__ y_out, int Kc, int T, int C, int L)
{
  const int c  = blockIdx.x;
  const int k  = blockIdx.y;
  const int l  = threadIdx.x;
  const int n  = l & 15;       // column (N) / A-row (M) index
  const int kh = l >> 4;       // lane half
  const int m  = n;            // A-operand row for this lane

  __shared__ float2 s_hk[16];       // gain * h[j]
  __shared__ float2 s_f[4][16];     // f_p[m]
  __shared__ float  s_x[2][8][16];  // double-buffered x: [buf][batch][time]

  // ---- per-channel coefficient preamble (lane 0, tiny serial) ----
  if (l == 0) {
    const float* ar = acoef + (size_t)k * 10;          // complex64 row (5 cplx)
    cpx g     = {ar[0], ar[1]};
    cpx aq[4] = {{ar[2],ar[3]},{ar[4],ar[5]},{ar[6],ar[7]},{ar[8],ar[9]}};
    cpx h[16];
    for (int j = 0; j < 16; ++j) {
      cpx acc = (j == 0) ? cpx{1.f, 0.f} : cpx{0.f, 0.f};
      for (int q = 1; q <= 4; ++q)
        if (j - q >= 0) acc = csub(acc, cmul(aq[q-1], h[j-q]));
      h[j] = acc;
      cpx hg = cmul(g, acc);
      s_hk[j] = make_float2(hg.r, hg.i);
    }
    for (int p = 1; p <= 4; ++p) {
      cpx f[16];
      for (int mm = 0; mm < 16; ++mm) {
        cpx acc = {0.f, 0.f};
        for (int q = 1; q <= 4; ++q) {
          if (mm - q >= 0)      acc = csub(acc, cmul(aq[q-1], f[mm-q]));
          else if (q - mm == p) acc = csub(acc, aq[q-1]);   // y[t0-p] seed
        }
        f[mm] = acc;
        s_f[p-1][mm] = make_float2(acc.r, acc.i);
      }
    }
  }
  __syncthreads();

  // ---- A operands: 4 K-chunks of the 16x16 Toeplitz (re & im) + 16x4 F ----
  // A 16x4 layout: VGPR0 = K 2*kh, VGPR1 = K 2*kh+1, M = lane&15.
  v2f Ar[4], Ai[4];
  #pragma unroll
  for (int cc = 0; cc < 4; ++cc) {
    const int kx = 4*cc + 2*kh, ky = kx + 1;
    float2 hx = (m - kx >= 0) ? s_hk[m - kx] : make_float2(0.f, 0.f);
    float2 hy = (m - ky >= 0) ? s_hk[m - ky] : make_float2(0.f, 0.f);
    Ar[cc].x = hx.x; Ar[cc].y = hy.x;
    Ai[cc].x = hx.y; Ai[cc].y = hy.y;
  }
  v2f Fr, Fi;
  { float2 f0 = s_f[2*kh][m], f1 = s_f[2*kh + 1][m];
    Fr.x = f0.x; Fr.y = f1.x; Fi.x = f0.y; Fi.y = f1.y; }

  // ---- incoming state (B-operand: rows 2*kh / 2*kh+1, col n) ----
  v2f S;
  if (STORE) {
    const float* sp = sin_ws + ((size_t)(k * C + c) * 4) * 16;
    S.x = sp[(2*kh    ) * 16 + n];
    S.y = sp[(2*kh + 1) * 16 + n];
  } else { S.x = 0.f; S.y = 0.f; }

  const int  bst  = l >> 2;          // staging batch
  const int  seg  = l & 3;           // staging 4-float segment
  const bool odd  = (n & 1) != 0;
  const int  bo   = n >> 1;          // output batch
  const int  part = n & 1;           // 0=re, 1=im

  const int nb = L / 16;
  const int chunk0 = c * L;
  const float* xstage = x + (size_t)bst * T + chunk0 + seg * 4;

  // prologue: stage block 0 into buffer 0
  { const float4 xv0 = *(const float4*)xstage;
    *(float4*)&s_x[0][bst][seg * 4] = xv0; }
  __syncthreads();

  for (int blk = 0; blk < nb; ++blk) {
    const int  t0       = chunk0 + blk * 16;
    const int  buf      = blk & 1;
    const bool havenext = (blk + 1) < nb;

    // issue next block's global load early; its s_wait lands after compute
    float4 xn;
    if (havenext) {
      xn = *(const float4*)(xstage + (blk + 1) * 16);
      if (blk + 2 < nb)
        __builtin_prefetch(xstage + (blk + 2) * 16, 0, 1);
    }

    v8f Y = {};
    // state contribution: Y += Fr*S + Fi*S~
    Y = WMMA_F32_16x16x4(Fr, S, Y);
    { float tx = __shfl_xor(S.x, 1);
      float ty = __shfl_xor(S.y, 1);
      v2f St; St.x = odd ? tx : -tx; St.y = odd ? ty : -ty;
      Y = WMMA_F32_16x16x4(Fi, St, Y); }
    // convolution: Y += sum_cc HKr*Xe + HKi*Xo   (K=16 as 4 chained K=4 wmmas)
    #pragma unroll
    for (int cc = 0; cc < 4; ++cc) {
      const float2 xv2 = *(const float2*)&s_x[buf][bo][4*cc + 2*kh];
      v2f Be, Bo;
      Be.x = odd ? 0.f : xv2.x; Be.y = odd ? 0.f : xv2.y;
      Bo.x = odd ? xv2.x : 0.f; Bo.y = odd ? xv2.y : 0.f;
      Y = WMMA_F32_16x16x4(Ar[cc], Be, Y);
      Y = WMMA_F32_16x16x4(Ai[cc], Bo, Y);
    }

    if (STORE) {
      // Y[r]: time t0 + r + 8*kh, batch bo, component part; streaming stores
      float* yp = y_out + (((size_t)bo * Kc + k) * T + t0 + 8*kh) * 2 + part;
      #pragma unroll
      for (int r = 0; r < 8; ++r)
        __builtin_nontemporal_store(Y[r], yp + (size_t)r * 2);
    }

    // next state = y[t0+15-row]; times 12..15 live in hi lanes of V4..V7
    { float a0 = __shfl(Y[7], 16 + n);
      float b0 = __shfl(Y[5], 16 + n);
      float a1 = __shfl(Y[6], 16 + n);
      float b1 = __shfl(Y[4], 16 + n);
      S.x = kh ? b0 : a0;     // row 2*kh
      S.y = kh ? b1 : a1; }   // row 2*kh+1

    // park next block's x into the other buffer (wait for load happens here)
    if (havenext)
      *(float4*)&s_x[buf ^ 1][bst][seg * 4] = xn;
    __syncthreads();
  }

  if (!STORE) {
    float* pp = p_ws + ((size_t)(k * C + c) * 4) * 16;
    pp[(2*kh    ) * 16 + n] = S.x;
    pp[(2*kh + 1) * 16 + n] = S.y;
  }
}

// ---- phase 2: per-channel sequential stitch s_{c+1} = G*s_c + p_c ----
__device__ __forceinline__ void cmat4_mul(const cpx A[4][4], const cpx B[4][4], cpx Cm[4][4]) {
  for (int i = 0; i < 4; ++i)
    for (int j = 0; j < 4; ++j) {
      cpx acc = {0.f, 0.f};
      for (int q = 0; q < 4; ++q) acc = cadd(acc, cmul(A[i][q], B[q][j]));
      Cm[i][j] = acc;
    }
}

__global__ void gt_stitch(const float* __restrict__ acoef,
                          const float* __restrict__ p_ws,
                          float* __restrict__ sin_ws,
                          int Kc, int C, int L)
{
  const int k = blockIdx.x;
  const int b = threadIdx.x;           // 0..7 (batch)
  const float* ar = acoef + (size_t)k * 10;
  cpx aq[4] = {{ar[2],ar[3]},{ar[4],ar[5]},{ar[6],ar[7]},{ar[8],ar[9]}};

  // companion matrix M; G = M^L by binary squaring
  cpx M[4][4];
  for (int i = 0; i < 4; ++i) for (int j = 0; j < 4; ++j) M[i][j] = {0.f, 0.f};
  for (int j = 0; j < 4; ++j) M[0][j] = {-aq[j].r, -aq[j].i};
  M[1][0] = {1.f,0.f}; M[2][1] = {1.f,0.f}; M[3][2] = {1.f,0.f};

  cpx G[4][4], Bp[4][4], Tm[4][4];
  for (int i = 0; i < 4; ++i) for (int j = 0; j < 4; ++j) {
    G[i][j]  = (i == j) ? cpx{1.f,0.f} : cpx{0.f,0.f};
    Bp[i][j] = M[i][j];
  }
  unsigned e = (unsigned)L;
  while (e) {
    if (e & 1) { cmat4_mul(G, Bp, Tm);
                 for (int i=0;i<4;++i) for (int j=0;j<4;++j) G[i][j]=Tm[i][j]; }
    e >>= 1;
    if (e) { cmat4_mul(Bp, Bp, Tm);
             for (int i=0;i<4;++i) for (int j=0;j<4;++j) Bp[i][j]=Tm[i][j]; }
  }

  cpx s[4] = {{0.f,0.f},{0.f,0.f},{0.f,0.f},{0.f,0.f}};
  for (int c = 0; c < C; ++c) {
    float* dst = sin_ws + ((size_t)(k * C + c) * 4) * 16;
    #pragma unroll
    for (int r = 0; r < 4; ++r) { dst[r*16 + 2*b] = s[r].r; dst[r*16 + 2*b + 1] = s[r].i; }
    const float* pp = p_ws + ((size_t)(k * C + c) * 4) * 16;
    cpx ns[4];
    #pragma unroll
    for (int r = 0; r < 4; ++r) {
      cpx acc = {pp[r*16 + 2*b], pp[r*16 + 2*b + 1]};
      for (int q = 0; q < 4; ++q) acc = cadd(acc, cmul(G[r][q], s[q]));
      ns[r] = acc;
    }
    #pragma unroll
    for (int r = 0; r < 4; ++r) s[r] = ns[r];
  }
}

extern "C" void kernel_launch(void* const* d_in, const int* in_sizes, int n_in,
                              void* d_out, int out_size, void* d_ws, size_t ws_size,
                              hipStream_t stream)
{
  const float* x = (const float*)d_in[0];
  const float* a = (const float*)d_in[1];   // complex64 rows viewed as floats
  float* y = (float*)d_out;

  const int B = 8;
  const int T = in_sizes[0] / B;
  int Kc = out_size / (2 * in_sizes[0]);    // out = B*Kc*T complex -> 2x floats
  if (Kc <= 0) Kc = out_size / in_sizes[0]; // fallback: out_size in complex
  if (Kc <= 0) Kc = 1;

  // chunk length selection: L multiple of 16; prefer many chunks (more waves)
  // as long as the two state arrays fit in the workspace.
  int L = T;
  const int cand[4] = {160, 320, 1600, 0};
  for (int ci = 0; ci < 3; ++ci) {
    const int Lc = cand[ci];
    if (Lc > 0 && T % Lc == 0) {
      const size_t need = (size_t)Kc * (T / Lc) * 64 * sizeof(float) * 2;
      if (need <= ws_size) { L = Lc; break; }
    }
  }
  int C = T / L;
  if ((size_t)Kc * C * 64 * sizeof(float) * 2 > ws_size) { L = T; C = 1; }

  float* p_ws   = (float*)d_ws;                    // chunk-final states
  float* sin_ws = p_ws + (size_t)Kc * C * 64;      // stitched incoming states

  dim3 grid(C, Kc);
  if (C > 1)
    gt_chunk<false><<<grid, 32, 0, stream>>>(x, a, nullptr, p_ws, nullptr, Kc, T, C, L);
  gt_stitch<<<Kc, 8, 0, stream>>>(a, p_ws, sin_ws, Kc, C, L);
  gt_chunk<true><<<grid, 32, 0, stream>>>(x, a, sin_ws, nullptr, y, Kc, T, C, L);
}